// GeoCov19HeteroGNN_7164005450261
// MI455X (gfx1250) — compile-verified
//
#include <hip/hip_runtime.h>

typedef __attribute__((ext_vector_type(2))) float v2f;
typedef __attribute__((ext_vector_type(8))) float v8f;

#define N_RET  200000
#define N_ORIG 100000
#define N_EDGE 800000
#define DF     128
#define OUTF   64

// Native f32 atomic add: relaxed + agent scope, result unused -> should lower to
// non-returning global_atomic_add_f32 (no CAS loop, no pre-op return traffic).
__device__ __forceinline__ void atomic_add_f32(float* p, float v) {
  (void)__hip_atomic_fetch_add(p, v, __ATOMIC_RELAXED, __HIP_MEMORY_SCOPE_AGENT);
}

// ---------------- zero fill (float4 grid-stride) ----------------
__global__ void k_zero(float4* __restrict__ p, long n4) {
  long i = (long)blockIdx.x * blockDim.x + threadIdx.x;
  const long s = (long)gridDim.x * blockDim.x;
  const float4 z = {0.f, 0.f, 0.f, 0.f};
  for (; i < n4; i += s) p[i] = z;
}

// ---------------- edge degree counts ----------------
__global__ void k_degrees(const int* __restrict__ ridx, const int* __restrict__ cidx,
                          float* __restrict__ degR, float* __restrict__ degO) {
  int e = blockIdx.x * blockDim.x + threadIdx.x;
  if (e < N_EDGE) {
    atomic_add_f32(&degR[ridx[e]], 1.0f);
    atomic_add_f32(&degO[cidx[e]], 1.0f);
  }
}

// ---------------- SAGE aggregate: agg[ret] += x_orig[orig] ----------------
// one wave (32 lanes) per edge, 4 feats per lane
__global__ void k_sage_aggregate(const float* __restrict__ xo,
                                 const int* __restrict__ sidx,   // orig_idx
                                 const int* __restrict__ didx,   // ret_idx
                                 float* __restrict__ agg) {
  const int e = blockIdx.x * 8 + (threadIdx.x >> 5);
  if (e >= N_EDGE) return;
  const int lane = threadIdx.x & 31;
  const int s = sidx[e], d = didx[e];
  const float4 v = *(const float4*)(xo + (size_t)s * DF + lane * 4);
  float* dst = agg + (size_t)d * DF + lane * 4;
  atomic_add_f32(dst + 0, v.x);
  atomic_add_f32(dst + 1, v.y);
  atomic_add_f32(dst + 2, v.z);
  atomic_add_f32(dst + 3, v.w);
}

// ---------------- GCN scatter: accum[c] += h[r] * norm(r,c) ----------------
__global__ void k_gcn_scatter(const float* __restrict__ h,
                              const int* __restrict__ ridx, const int* __restrict__ cidx,
                              const float* __restrict__ degR, const float* __restrict__ degO,
                              float* __restrict__ accum) {
  const int e = blockIdx.x * 8 + (threadIdx.x >> 5);
  if (e >= N_EDGE) return;
  const int lane = threadIdx.x & 31;
  const int r = ridx[e], c = cidx[e];
  const float ds = degR[r], dd = degO[c];
  const float norm = (ds > 0.f ? rsqrtf(ds) : 0.f) * (dd > 0.f ? rsqrtf(dd) : 0.f);
  const float4 v = *(const float4*)(h + (size_t)r * DF + lane * 4);
  float* dst = accum + (size_t)c * DF + lane * 4;
  atomic_add_f32(dst + 0, v.x * norm);
  atomic_add_f32(dst + 1, v.y * norm);
  atomic_add_f32(dst + 2, v.z * norm);
  atomic_add_f32(dst + 3, v.w * norm);
}

// ---------------- SAGE fused dual GEMM + bias + relu (WMMA f32) ----------------
// out = relu((agg/max(degR,1)) @ Wl + bl + xr @ Wr), in-place over agg buffer.
// 256 threads = 8 waves; wave w handles output cols [16w, 16w+16); 16 rows/block.
__global__ __launch_bounds__(256)
void k_sage_gemm(const float* __restrict__ agg, const float* __restrict__ xr,
                 const float* __restrict__ degR,
                 const float* __restrict__ Wl, const float* __restrict__ bl,
                 const float* __restrict__ Wr, float* __restrict__ out) {
  __shared__ float sA[16][132];
  __shared__ float sX[16][132];
  const int rb = blockIdx.x * 16;
  const int t = threadIdx.x;
  for (int i = t; i < 16 * 32; i += 256) {
    const int row = i >> 5;
    const int c4 = (i & 31) << 2;
    const float inv = 1.0f / fmaxf(degR[rb + row], 1.0f);
    const float4 a = *(const float4*)(agg + (size_t)(rb + row) * DF + c4);
    const float4 x = *(const float4*)(xr + (size_t)(rb + row) * DF + c4);
    sA[row][c4 + 0] = a.x * inv; sA[row][c4 + 1] = a.y * inv;
    sA[row][c4 + 2] = a.z * inv; sA[row][c4 + 3] = a.w * inv;
    sX[row][c4 + 0] = x.x; sX[row][c4 + 1] = x.y;
    sX[row][c4 + 2] = x.z; sX[row][c4 + 3] = x.w;
  }
  __syncthreads();

  const int lane = t & 31;
  const int half = lane >> 4;   // K sub-block select per ISA A/B layout
  const int l16 = lane & 15;    // M (A rows) / N (B cols)
  const int n0 = (t >> 5) * 16;

  v8f acc = {0.f, 0.f, 0.f, 0.f, 0.f, 0.f, 0.f, 0.f};
#pragma unroll 8
  for (int k0 = 0; k0 < DF; k0 += 4) {
    const int kk = k0 + 2 * half;
    v2f a1; a1.x = sA[l16][kk]; a1.y = sA[l16][kk + 1];
    v2f b1; b1.x = Wl[(size_t)kk * DF + n0 + l16];
    b1.y = Wl[(size_t)(kk + 1) * DF + n0 + l16];
    acc = __builtin_amdgcn_wmma_f32_16x16x4_f32(false, a1, false, b1,
                                                (short)0, acc, false, false);
    v2f a2; a2.x = sX[l16][kk]; a2.y = sX[l16][kk + 1];
    v2f b2; b2.x = Wr[(size_t)kk * DF + n0 + l16];
    b2.y = Wr[(size_t)(kk + 1) * DF + n0 + l16];
    acc = __builtin_amdgcn_wmma_f32_16x16x4_f32(false, a2, false, b2,
                                                (short)0, acc, false, false);
  }
  const float bias = bl[n0 + l16];
#pragma unroll
  for (int j = 0; j < 8; ++j) {
    const float v = fmaxf(acc[j] + bias, 0.f);
    out[(size_t)(rb + j + 8 * half) * DF + n0 + l16] = v;
  }
}

// ---------------- plain GEMM: out = A @ W (128x128), in-place over A buffer ----------------
__global__ __launch_bounds__(256)
void k_gemm_plain(const float* __restrict__ A, const float* __restrict__ W,
                  float* __restrict__ out) {
  __shared__ float sA[16][132];
  const int rb = blockIdx.x * 16;
  const int t = threadIdx.x;
  for (int i = t; i < 16 * 32; i += 256) {
    const int row = i >> 5;
    const int c4 = (i & 31) << 2;
    const float4 a = *(const float4*)(A + (size_t)(rb + row) * DF + c4);
    sA[row][c4 + 0] = a.x; sA[row][c4 + 1] = a.y;
    sA[row][c4 + 2] = a.z; sA[row][c4 + 3] = a.w;
  }
  __syncthreads();

  const int lane = t & 31;
  const int half = lane >> 4;
  const int l16 = lane & 15;
  const int n0 = (t >> 5) * 16;

  v8f acc = {0.f, 0.f, 0.f, 0.f, 0.f, 0.f, 0.f, 0.f};
#pragma unroll 8
  for (int k0 = 0; k0 < DF; k0 += 4) {
    const int kk = k0 + 2 * half;
    v2f a; a.x = sA[l16][kk]; a.y = sA[l16][kk + 1];
    v2f b; b.x = W[(size_t)kk * DF + n0 + l16];
    b.y = W[(size_t)(kk + 1) * DF + n0 + l16];
    acc = __builtin_amdgcn_wmma_f32_16x16x4_f32(false, a, false, b,
                                                (short)0, acc, false, false);
  }
#pragma unroll
  for (int j = 0; j < 8; ++j)
    out[(size_t)(rb + j + 8 * half) * DF + n0 + l16] = acc[j];
}

// ---------------- readout: out = relu(accum + b_g2) @ W_out + b_out ----------------
// 128 threads = 4 waves -> N = 64 cols. 16 rows/block.
__global__ __launch_bounds__(128)
void k_out_gemm(const float* __restrict__ accum, const float* __restrict__ bg,
                const float* __restrict__ Wo, const float* __restrict__ bo,
                float* __restrict__ out) {
  __shared__ float sA[16][132];
  const int rb = blockIdx.x * 16;
  const int t = threadIdx.x;
  for (int i = t; i < 16 * 32; i += 128) {
    const int row = i >> 5;
    const int c4 = (i & 31) << 2;
    const float4 a = *(const float4*)(accum + (size_t)(rb + row) * DF + c4);
    const float4 b = *(const float4*)(bg + c4);
    sA[row][c4 + 0] = fmaxf(a.x + b.x, 0.f);
    sA[row][c4 + 1] = fmaxf(a.y + b.y, 0.f);
    sA[row][c4 + 2] = fmaxf(a.z + b.z, 0.f);
    sA[row][c4 + 3] = fmaxf(a.w + b.w, 0.f);
  }
  __syncthreads();

  const int lane = t & 31;
  const int half = lane >> 4;
  const int l16 = lane & 15;
  const int n0 = (t >> 5) * 16;

  v8f acc = {0.f, 0.f, 0.f, 0.f, 0.f, 0.f, 0.f, 0.f};
#pragma unroll 8
  for (int k0 = 0; k0 < DF; k0 += 4) {
    const int kk = k0 + 2 * half;
    v2f a; a.x = sA[l16][kk]; a.y = sA[l16][kk + 1];
    v2f b; b.x = Wo[(size_t)kk * OUTF + n0 + l16];
    b.y = Wo[(size_t)(kk + 1) * OUTF + n0 + l16];
    acc = __builtin_amdgcn_wmma_f32_16x16x4_f32(false, a, false, b,
                                                (short)0, acc, false, false);
  }
  const float bias = bo[n0 + l16];
#pragma unroll
  for (int j = 0; j < 8; ++j)
    out[(size_t)(rb + j + 8 * half) * OUTF + n0 + l16] = acc[j] + bias;
}

extern "C" void kernel_launch(void* const* d_in, const int* in_sizes, int n_in,
                              void* d_out, int out_size, void* d_ws, size_t ws_size,
                              hipStream_t stream) {
  const float* x_ret    = (const float*)d_in[0];
  const float* x_orig   = (const float*)d_in[1];
  const int*   ret_idx  = (const int*)d_in[2];
  const int*   orig_idx = (const int*)d_in[3];
  // d_in[4]=W_g1, d_in[5]=b_g1 unused (o1 is dead); d_in[11..13]=Wl2,bl2,Wr2 unused (r2 dead)
  const float* Wl1   = (const float*)d_in[6];
  const float* bl1   = (const float*)d_in[7];
  const float* Wr1   = (const float*)d_in[8];
  const float* W_g2  = (const float*)d_in[9];
  const float* b_g2  = (const float*)d_in[10];
  const float* W_out = (const float*)d_in[14];
  const float* b_out = (const float*)d_in[15];
  float* out = (float*)d_out;

  float* buf1  = (float*)d_ws;                      // N_RET*DF : agg -> r1 -> h2
  float* accum = buf1 + (size_t)N_RET * DF;         // N_ORIG*DF
  float* degR  = accum + (size_t)N_ORIG * DF;       // N_RET
  float* degO  = degR + N_RET;                      // N_ORIG
  const long ztotal = (long)N_RET * DF + (long)N_ORIG * DF + N_RET + N_ORIG;

  k_zero<<<2048, 256, 0, stream>>>((float4*)d_ws, ztotal / 4);
  k_degrees<<<(N_EDGE + 255) / 256, 256, 0, stream>>>(ret_idx, orig_idx, degR, degO);
  k_sage_aggregate<<<N_EDGE / 8, 256, 0, stream>>>(x_orig, orig_idx, ret_idx, buf1);
  k_sage_gemm<<<N_RET / 16, 256, 0, stream>>>(buf1, x_ret, degR, Wl1, bl1, Wr1, buf1);
  k_gemm_plain<<<N_RET / 16, 256, 0, stream>>>(buf1, W_g2, buf1);
  k_gcn_scatter<<<N_EDGE / 8, 256, 0, stream>>>(buf1, ret_idx, orig_idx, degR, degO, accum);
  k_out_gemm<<<N_ORIG / 16, 128, 0, stream>>>(accum, b_g2, W_out, b_out, out);
}